// MeanPooling_34815004901952
// MI455X (gfx1250) — compile-verified
//
#include <hip/hip_runtime.h>
#include <cstdint>
#include <cstddef>

#define FEAT      128      // D
#define NGR       4096     // num_graphs (fixed by reference)
#define WAVES     8        // waves per workgroup (wave32)
#define CHUNK     4        // rows per async double-buffer chunk per wave

// ---------------------------------------------------------------------------
// Kernel 1: per-graph row ranges via binary search on the sorted batch array.
// starts[g] = first row index with batch[row] >= g, for g in [0, NGR].
// ---------------------------------------------------------------------------
__global__ void k_bounds(const int* __restrict__ batch, int n,
                         int* __restrict__ starts) {
    int g = blockIdx.x * blockDim.x + threadIdx.x;
    if (g > NGR) return;
    int lo = 0, hi = n;
    while (lo < hi) {
        int mid = (lo + hi) >> 1;
        if (batch[mid] < g) lo = mid + 1; else hi = mid;
    }
    starts[g] = lo;
}

// ---------------------------------------------------------------------------
// Kernel 2: one workgroup per graph. Rows stream through a double-buffered
// LDS stage filled with CDNA5 async global->LDS copies (ASYNCcnt-tracked),
// marked non-temporal: x is 512 MB read exactly once (> 192 MB L2), so lines
// must not displace reusable state. Lane l owns feature columns [4l, 4l+4)
// as a float4; one async b128 instruction moves one full 512 B row.
// ---------------------------------------------------------------------------
__global__ __launch_bounds__(WAVES * 32)
void k_pool(const float* __restrict__ x,
            const int* __restrict__ starts,
            float* __restrict__ out) {
    __shared__ float4 stage[WAVES][2][CHUNK][32];   // 32 KB async staging
    __shared__ float4 partial[WAVES][32];           // 4 KB cross-wave combine

    const int g    = blockIdx.x;
    const int wave = threadIdx.x >> 5;
    const int lane = threadIdx.x & 31;

    const int s   = starts[g];
    const int e   = starts[g + 1];
    const int cnt = e - s;

    // Contiguous slice of this graph's rows for this wave.
    const int q   = cnt / WAVES;
    const int rem = cnt % WAVES;
    const int my_start = s + wave * q + (wave < rem ? wave : rem);
    const int my_len   = q + (wave < rem ? 1 : 0);

    const int nchunks = my_len / CHUNK;
    const int tail    = my_len - nchunks * CHUNK;

    float4 acc = make_float4(0.f, 0.f, 0.f, 0.f);

    const uint64_t xbase = (uint64_t)(uintptr_t)x + (uint64_t)lane * 16u;

    if (nchunks > 0) {
        // Prime the pipeline: issue async fills for chunk 0 into buffer 0.
        {
            uint64_t gbase = xbase + (uint64_t)my_start * (FEAT * 4);
#pragma unroll
            for (int r = 0; r < CHUNK; ++r) {
                uint32_t laddr = (uint32_t)(uintptr_t)&stage[wave][0][r][lane];
                uint64_t gaddr = gbase + (uint64_t)r * (FEAT * 4);
                asm volatile(
                    "global_load_async_to_lds_b128 %0, %1, off th:TH_LOAD_NT"
                    :: "v"(laddr), "v"(gaddr) : "memory");
            }
        }
        for (int c = 0; c < nchunks; ++c) {
            const int buf = c & 1;
            if (c + 1 < nchunks) {
                const int nb = buf ^ 1;
                // Buffer nb was last *read* in iteration c-1; make sure those
                // LDS reads retired before the async engine overwrites it.
                asm volatile("s_wait_dscnt 0" ::: "memory");
                uint64_t gbase = xbase +
                    (uint64_t)(my_start + (c + 1) * CHUNK) * (FEAT * 4);
#pragma unroll
                for (int r = 0; r < CHUNK; ++r) {
                    uint32_t laddr =
                        (uint32_t)(uintptr_t)&stage[wave][nb][r][lane];
                    uint64_t gaddr = gbase + (uint64_t)r * (FEAT * 4);
                    asm volatile(
                        "global_load_async_to_lds_b128 %0, %1, off th:TH_LOAD_NT"
                        :: "v"(laddr), "v"(gaddr) : "memory");
                }
                // Allow the CHUNK just-issued fills to stay outstanding; the
                // current chunk's CHUNK fills (issued earlier, in-order) done.
                asm volatile("s_wait_asynccnt 4" ::: "memory");
            } else {
                asm volatile("s_wait_asynccnt 0" ::: "memory");
            }
#pragma unroll
            for (int r = 0; r < CHUNK; ++r) {
                float4 v = stage[wave][buf][r][lane];
                acc.x += v.x; acc.y += v.y; acc.z += v.z; acc.w += v.w;
            }
        }
    }

    // Tail rows (< CHUNK): direct coalesced global loads.
    {
        const float4* xv = (const float4*)x;
        const int base = my_start + nchunks * CHUNK;
        for (int r = 0; r < tail; ++r) {
            float4 v = xv[(size_t)(base + r) * (FEAT / 4) + lane];
            acc.x += v.x; acc.y += v.y; acc.z += v.z; acc.w += v.w;
        }
    }

    partial[wave][lane] = acc;
    __syncthreads();

    if (wave == 0) {
        float4 t = partial[0][lane];
#pragma unroll
        for (int w2 = 1; w2 < WAVES; ++w2) {
            float4 p = partial[w2][lane];
            t.x += p.x; t.y += p.y; t.z += p.z; t.w += p.w;
        }
        const float inv = 1.0f / (float)(cnt > 0 ? cnt : 1);
        float4 o = make_float4(t.x * inv, t.y * inv, t.z * inv, t.w * inv);
        ((float4*)out)[(size_t)g * (FEAT / 4) + lane] = o;
    }
}

// ---------------------------------------------------------------------------
// Host entry. Inputs: d_in[0]=x (N*128 f32), d_in[1]=batch (N int, sorted),
// d_in[2]=num_graphs scalar. Output: 4096*128 f32 means.
// ---------------------------------------------------------------------------
extern "C" void kernel_launch(void* const* d_in, const int* in_sizes, int n_in,
                              void* d_out, int out_size, void* d_ws, size_t ws_size,
                              hipStream_t stream) {
    const float* x     = (const float*)d_in[0];
    const int*   batch = (const int*)d_in[1];
    const int    n     = in_sizes[1];          // number of rows
    int* starts        = (int*)d_ws;           // (NGR + 1) ints of scratch

    k_bounds<<<(NGR + 1 + 255) / 256, 256, 0, stream>>>(batch, n, starts);
    k_pool<<<NGR, WAVES * 32, 0, stream>>>(x, starts, (float*)d_out);
}